// KGVAE_77996606095487
// MI455X (gfx1250) — compile-verified
//
#include <hip/hip_runtime.h>
#include <hip/hip_bf16.h>
#include <math.h>
#include <stdint.h>

// ---------------------------------------------------------------------------
// KGVAE RelGraphConv (bdd) x2 + reparameterize for MI455X (gfx1250, wave32)
//
// - Self-loop GEMMs: V_WMMA_F32_16X16X4_F32 (f32 matrix pipe, exact precision)
// - Edge messages: gather + guaranteed-hardware global_atomic_add_f32 scatter
// - Whole working set (~60MB) is L2-resident on the 192MB L2; bound by L2
//   atomic/bandwidth throughput, not HBM.
// ---------------------------------------------------------------------------

typedef __attribute__((ext_vector_type(2))) float v2f;
typedef __attribute__((ext_vector_type(8))) float v8f;

#define H_DIM 64

// Hardware f32 atomic add (no-return form, STOREcnt-tracked; drained by the
// implicit wait-idle at s_endpgm). Avoids any possibility of a CAS-loop
// expansion from generic atomicAdd.
__device__ __forceinline__ void atomic_add_f32(float* p, float v) {
  asm volatile("global_atomic_add_f32 %0, %1, off"
               :
               : "v"((unsigned long long)(uintptr_t)p), "v"(v)
               : "memory");
}

// ---------------------------------------------------------------------------
// C[N,OUT] = gather(A, ids)[N,64] @ W[64,OUT] + bias[OUT]
// One wave32 per 16x16 output tile; K=64 as 16 steps of V_WMMA_F32_16X16X4_F32.
//
// ISA layouts (CDNA5 ISA §7.12.2):
//   A 16x4 f32 : lanes 0-15 -> M=lane,    v0=K0, v1=K1
//                lanes 16-31-> M=lane-16, v0=K2, v1=K3
//   B 4x16 f32 : lanes 0-15 -> N=lane,    v0=K0, v1=K1  (symmetric to A)
//                lanes 16-31-> N=lane-16, v0=K2, v1=K3
//   C/D 16x16  : VGPR r: lanes 0-15 -> M=r, lanes 16-31 -> M=r+8; N=lane%16
// ---------------------------------------------------------------------------
template <int OUT>
__global__ __launch_bounds__(256) void selfloop_gemm_wmma(
    const float* __restrict__ A, const int* __restrict__ ids,
    const float* __restrict__ W, const float* __restrict__ bias,
    float* __restrict__ C, int nrows) {
  const int lane = threadIdx.x & 31;
  const int wave = (int)((blockIdx.x * blockDim.x + threadIdx.x) >> 5);
  const int tiles_n = OUT / 16;
  const int tile_m = wave / tiles_n;
  const int tile_n = wave - tile_m * tiles_n;
  const int row0 = tile_m * 16;
  if (row0 >= nrows) return;  // wave-uniform: EXEC stays all-ones for WMMA

  const int half = lane >> 4;   // 0: lanes 0-15, 1: lanes 16-31
  const int l16  = lane & 15;
  const int col  = tile_n * 16 + l16;

  int arow = row0 + l16;
  if (arow >= nrows) arow = nrows - 1;          // clamp; stores are guarded
  const int grow = ids ? ids[arow] : arow;      // embedding-lookup gather
  const float* __restrict__ arowp = A + (size_t)grow * H_DIM;

  const float bcol = bias[col];
  v8f acc;
#pragma unroll
  for (int r = 0; r < 8; ++r) acc[r] = bcol;    // C initialized with bias

#pragma unroll
  for (int k = 0; k < H_DIM; k += 4) {
    // A fragment: this lane holds A[row][k+2h], A[row][k+2h+1]
    v2f a = *(const v2f*)(arowp + k + 2 * half);
    // B fragment: this lane holds W[k+2h][col], W[k+2h+1][col]
    v2f b;
    b.x = W[(k + 2 * half) * OUT + col];
    b.y = W[(k + 2 * half + 1) * OUT + col];
    acc = __builtin_amdgcn_wmma_f32_16x16x4_f32(
        /*neg_a=*/false, a, /*neg_b=*/false, b,
        /*c_mod=*/(short)0, acc, /*reuse_a=*/false, /*reuse_b=*/false);
  }

#pragma unroll
  for (int r = 0; r < 8; ++r) {
    const int row = row0 + r + 8 * half;
    if (row < nrows) C[(size_t)row * OUT + col] = acc[r];
  }
}

// ---------------------------------------------------------------------------
// Layer-1 edge messages: out_feat=64, blocks of 8x8.
// One thread per (edge, out-elem). Each edge spans exactly 2 full wave32s, so
// the edge index is wave-uniform -> readfirstlane to force SGPR scalar loads
// of src/dst/rel/norm and saddr-based addressing for the x/w vector loads.
// msg[e, b*8+j] = sum_i feat[src[e]][b*8+i] * w1[rel[e]][b*64 + i*8 + j]
// ---------------------------------------------------------------------------
__global__ __launch_bounds__(256) void edge_msg_l1(
    const float* __restrict__ emb, const int* __restrict__ ids,
    const int* __restrict__ src, const int* __restrict__ dst,
    const int* __restrict__ rel, const float* __restrict__ norm,
    const float* __restrict__ w1, float* __restrict__ agg, int nE) {
  const int idx = blockIdx.x * blockDim.x + threadIdx.x;
  if (idx >= nE * 64) return;
  const int e = __builtin_amdgcn_readfirstlane(idx >> 6);  // wave-uniform
  const int o = idx & 63;
  const int b = o >> 3, j = o & 7;
  const int s = ids[src[e]];
  const int d = dst[e];
  const int r = rel[e];
  const float nm = norm[e];
  const float4* __restrict__ x4 = (const float4*)(emb + (size_t)s * 64 + b * 8);
  const float* __restrict__ w = w1 + (size_t)r * 512 + b * 64 + j;
  const float4 xa = x4[0], xb = x4[1];
  float acc = xa.x * w[0]  + xa.y * w[8]  + xa.z * w[16] + xa.w * w[24]
            + xb.x * w[32] + xb.y * w[40] + xb.z * w[48] + xb.w * w[56];
  atomic_add_f32(agg + (size_t)d * 64 + o, acc * nm);
}

// ---------------------------------------------------------------------------
// Layer-2 edge messages: out_feat=128, blocks of 8x16. One edge = 4 wave32s.
// ---------------------------------------------------------------------------
__global__ __launch_bounds__(256) void edge_msg_l2(
    const float* __restrict__ h1, const int* __restrict__ src,
    const int* __restrict__ dst, const int* __restrict__ rel,
    const float* __restrict__ norm, const float* __restrict__ w2,
    float* __restrict__ agg, int nE) {
  const int idx = blockIdx.x * blockDim.x + threadIdx.x;
  if (idx >= nE * 128) return;
  const int e = __builtin_amdgcn_readfirstlane(idx >> 7);  // wave-uniform
  const int o = idx & 127;
  const int b = o >> 4, j = o & 15;
  const int s = src[e];
  const int d = dst[e];
  const int r = rel[e];
  const float nm = norm[e];
  const float4* __restrict__ x4 = (const float4*)(h1 + (size_t)s * 64 + b * 8);
  const float* __restrict__ w = w2 + (size_t)r * 1024 + b * 128 + j;
  const float4 xa = x4[0], xb = x4[1];
  float acc = xa.x * w[0]  + xa.y * w[16] + xa.z * w[32]  + xa.w * w[48]
            + xb.x * w[64] + xb.y * w[80] + xb.z * w[96]  + xb.w * w[112];
  atomic_add_f32(agg + (size_t)d * 128 + o, acc * nm);
}

__global__ __launch_bounds__(256) void relu_inplace(float* __restrict__ x, int n) {
  const int i = blockIdx.x * blockDim.x + threadIdx.x;
  if (i < n) x[i] = fmaxf(x[i], 0.0f);
}

// out = m + sqrt(softplus(raw_v) + 1e-8) * eps   (h2 split along last dim)
__global__ __launch_bounds__(256) void reparam(
    const float* __restrict__ h2,   // [N,128]
    const float* __restrict__ eps,  // [N,64]
    float* __restrict__ out, int n, int half) {
  const int i = blockIdx.x * blockDim.x + threadIdx.x;
  if (i >= n) return;
  const int row = i / half, k = i - row * half;
  const float m  = h2[(size_t)row * (2 * half) + k];
  const float rv = h2[(size_t)row * (2 * half) + half + k];
  // numerically stable softplus: max(x,0) + log1p(exp(-|x|))
  const float sp = fmaxf(rv, 0.0f) + log1pf(expf(-fabsf(rv)));
  out[i] = m + sqrtf(sp + 1e-8f) * eps[i];
}

extern "C" void kernel_launch(void* const* d_in, const int* in_sizes, int n_in,
                              void* d_out, int out_size, void* d_ws, size_t ws_size,
                              hipStream_t stream) {
  const float* emb  = (const float*)d_in[0];   // [N,64]
  const float* norm = (const float*)d_in[1];   // [E,1]
  const float* eps  = (const float*)d_in[2];   // [N,64]
  const float* w1   = (const float*)d_in[3];   // [R,512]
  const float* lw1  = (const float*)d_in[4];   // [64,64]
  const float* b1   = (const float*)d_in[5];   // [64]
  const float* w2   = (const float*)d_in[6];   // [R,1024]
  const float* lw2  = (const float*)d_in[7];   // [64,128]
  const float* b2   = (const float*)d_in[8];   // [128]
  const int* h_ids  = (const int*)d_in[9];     // [N]
  const int* src    = (const int*)d_in[10];    // [E]
  const int* dst    = (const int*)d_in[11];    // [E]
  const int* rel    = (const int*)d_in[12];    // [E]

  const int N = in_sizes[0] / 64;
  const int E = in_sizes[10];

  float* tmp1 = (float*)d_ws;                  // [N,64]  (becomes h1 in-place)
  float* tmp2 = tmp1 + (size_t)N * 64;         // [N,128]

  const int threads = 256;

  // 1) tmp1 = emb[h_ids] @ loop_w1 + b1              (WMMA f32)
  {
    const int waves = ((N + 15) / 16) * (64 / 16);
    const int blocks = (waves * 32 + threads - 1) / threads;
    selfloop_gemm_wmma<64><<<blocks, threads, 0, stream>>>(emb, h_ids, lw1, b1, tmp1, N);
  }
  // 2) scatter-add layer-1 edge messages into tmp1
  {
    const int total = E * 64;
    edge_msg_l1<<<(total + threads - 1) / threads, threads, 0, stream>>>(
        emb, h_ids, src, dst, rel, norm, w1, tmp1, E);
  }
  // 3) h1 = relu(tmp1) in place
  relu_inplace<<<(N * 64 + threads - 1) / threads, threads, 0, stream>>>(tmp1, N * 64);

  // 4) tmp2 = h1 @ loop_w2 + b2                      (WMMA f32)
  {
    const int waves = ((N + 15) / 16) * (128 / 16);
    const int blocks = (waves * 32 + threads - 1) / threads;
    selfloop_gemm_wmma<128><<<blocks, threads, 0, stream>>>(tmp1, nullptr, lw2, b2, tmp2, N);
  }
  // 5) scatter-add layer-2 edge messages into tmp2
  {
    const int total = E * 128;
    edge_msg_l2<<<(total + threads - 1) / threads, threads, 0, stream>>>(
        tmp1, src, dst, rel, norm, w2, tmp2, E);
  }
  // 6) reparameterize -> out [N,64]
  reparam<<<(N * 64 + threads - 1) / threads, threads, 0, stream>>>(tmp2, eps, (float*)d_out, N * 64, 64);
}